// ProteinConv_66915590472255
// MI455X (gfx1250) — compile-verified
//
#include <hip/hip_runtime.h>
#include <hip/hip_bf16.h>
#include <math.h>

typedef __attribute__((ext_vector_type(2))) float v2f;
typedef __attribute__((ext_vector_type(8))) float v8f;

#define N_B   4
#define H_IN  1024
#define H_P   1018      // (1024 - 7) + 1
#define H_PAD 1024      // padded rows, rows [1018,1024) are zero
#define D_D   64
#define F_F   32
#define C_C   8
#define W_W   7
#define SSTRIDE 1034    // 16-row score buffer stride (floats); conflict-free padding

// ---------------------------------------------------------------------------
// Repack W: (f,c,d,j) -> Wt[((d*8+c)*7+j)*32 + f]   (f fastest => float4 loads)
// ---------------------------------------------------------------------------
__global__ void repack_W_kernel(const float* __restrict__ W, float* __restrict__ Wt) {
    int idx = blockIdx.x * blockDim.x + threadIdx.x;
    if (idx >= F_F * C_C * D_D * W_W) return;
    int f  = idx & 31;
    int t  = idx >> 5;          // (d*8+c)*7 + j
    int j  = t % 7;
    int cd = t / 7;             // d*8 + c
    int c  = cd & 7;
    int d  = cd >> 3;
    Wt[idx] = W[((f * C_C + c) * D_D + d) * W_W + j];
}

// ---------------------------------------------------------------------------
// Projection: Q[n][f][h][d] = sum_{c,j} X[n][h+j][c][d] * W[f][c][d][j]
// thread = (n,h,d); h in [1018,1024) writes zeros (padding for WMMA K loops).
// ---------------------------------------------------------------------------
__global__ __launch_bounds__(256)
void project_kernel(const float* __restrict__ X, const float* __restrict__ Wt,
                    float* __restrict__ Q) {
    int idx = blockIdx.x * blockDim.x + threadIdx.x;   // N*H_IN*D_D threads
    int d = idx & 63;
    int h = (idx >> 6) & 1023;
    int n = idx >> 16;

    float acc[F_F];
#pragma unroll
    for (int f = 0; f < F_F; ++f) acc[f] = 0.0f;

    if (h < H_P) {
        for (int c = 0; c < C_C; ++c) {
#pragma unroll
            for (int j = 0; j < W_W; ++j) {
                float xv = X[((n * H_IN + h + j) * C_C + c) * D_D + d];
                const float4* wp =
                    (const float4*)(Wt + ((d * C_C + c) * W_W + j) * F_F);
#pragma unroll
                for (int f4 = 0; f4 < 8; ++f4) {
                    float4 w = wp[f4];
                    acc[f4 * 4 + 0] = fmaf(xv, w.x, acc[f4 * 4 + 0]);
                    acc[f4 * 4 + 1] = fmaf(xv, w.y, acc[f4 * 4 + 1]);
                    acc[f4 * 4 + 2] = fmaf(xv, w.z, acc[f4 * 4 + 2]);
                    acc[f4 * 4 + 3] = fmaf(xv, w.w, acc[f4 * 4 + 3]);
                }
            }
        }
    }
#pragma unroll
    for (int f = 0; f < F_F; ++f)
        Q[((long)(n * F_F + f) * H_PAD + h) * D_D + d] = acc[f];
}

// ---------------------------------------------------------------------------
// Attention pass: P = softmax(q1 q2^T / 1018) @ V for one (n, f, 16-row block).
// 4 waves / 128 threads. Dynamic LDS: S[16][SSTRIDE] + red[128] + rowmax[16]
// + rowinv[16].
// pass==0: store raw P to Out (padded [slice][1024][64] layout, pad rows = 0)
// pass==1: store tanh(P) transposed to Out[((n*1018+h)*32+f)*64+d]
// ---------------------------------------------------------------------------
__global__ __launch_bounds__(128)
void attn_kernel(const float* __restrict__ Q1, const float* __restrict__ Q2,
                 const float* __restrict__ V, float* __restrict__ Out, int pass) {
    extern __shared__ float smem[];
    float* Sbuf   = smem;                    // 16 * SSTRIDE
    float* red    = Sbuf + 16 * SSTRIDE;     // 128
    float* rowmax = red + 128;               // 16
    float* rowinv = rowmax + 16;             // 16

    const int tid  = threadIdx.x;
    const int wave = tid >> 5;
    const int lane = tid & 31;
    const int half = lane >> 4;
    const int l    = lane & 15;

    const int h0 = blockIdx.x * 16;
    const int f  = blockIdx.y;
    const int n  = blockIdx.z;
    const long slice = (long)(n * F_F + f) * (H_PAD * D_D);
    const float* q1 = Q1 + slice;
    const float* q2 = Q2 + slice;
    const float* vp = V + slice;

    // Preload q1 A-fragments for the whole row block: a[s] = q1[h0+l][4s+2*half .. +1]
    v2f aFrag[16];
    {
        const float* rowp = q1 + (h0 + l) * D_D + 2 * half;
#pragma unroll
        for (int s = 0; s < 16; ++s)
            aFrag[s] = *(const v2f*)(rowp + 4 * s);
    }

    // ---- Phase 1: score tiles S[16 x 1024] via f32 WMMA ----
    for (int gt = wave; gt < 64; gt += 4) {
        const int g0 = gt * 16;
        v8f acc = {0.f, 0.f, 0.f, 0.f, 0.f, 0.f, 0.f, 0.f};
        const float* bp = q2 + (g0 + l) * D_D + 2 * half;
#pragma unroll
        for (int s = 0; s < 16; ++s) {
            v2f b = *(const v2f*)(bp + 4 * s);
            acc = __builtin_amdgcn_wmma_f32_16x16x4_f32(
                false, aFrag[s], false, b, (short)0, acc, false, false);
        }
#pragma unroll
        for (int r = 0; r < 8; ++r)
            Sbuf[(r + 8 * half) * SSTRIDE + g0 + l] = acc[r];
    }
    __syncthreads();

    // ---- Phase 2: softmax over g (scale 1/1018 folded into exp) ----
    const int row = tid & 15;
    const int grp = tid >> 4;                // 0..7
    float* srow = Sbuf + row * SSTRIDE;

    float m = -INFINITY;
    for (int g = grp; g < H_P; g += 8) m = fmaxf(m, srow[g]);
    red[row * 8 + grp] = m;
    __syncthreads();
    if (tid < 16) {
        float mm = red[tid * 8];
#pragma unroll
        for (int k = 1; k < 8; ++k) mm = fmaxf(mm, red[tid * 8 + k]);
        rowmax[tid] = mm;
    }
    __syncthreads();

    const float invScale = 1.0f / 1018.0f;
    const float mrow = rowmax[row];
    float ssum = 0.0f;
    for (int g = grp; g < H_P; g += 8) {
        float e = __expf((srow[g] - mrow) * invScale);
        srow[g] = e;
        ssum += e;
    }
    red[row * 8 + grp] = ssum;
    if (tid < 16) {                          // zero padded columns 1018..1023
#pragma unroll
        for (int g = H_P; g < H_PAD; ++g) Sbuf[tid * SSTRIDE + g] = 0.0f;
    }
    __syncthreads();
    if (tid < 16) {
        float s = 0.0f;
#pragma unroll
        for (int k = 0; k < 8; ++k) s += red[tid * 8 + k];
        rowinv[tid] = 1.0f / s;
    }
    __syncthreads();

    // ---- Phase 3: P_block = E @ V, one 16-wide d-tile per wave ----
    const int n0 = wave * 16;
    v8f acc = {0.f, 0.f, 0.f, 0.f, 0.f, 0.f, 0.f, 0.f};
    const float* vb = vp + n0 + l;
    const float* arow = Sbuf + l * SSTRIDE + 2 * half;
#pragma unroll 8
    for (int s = 0; s < 256; ++s) {
        v2f a = *(const v2f*)(arow + 4 * s);
        const int k = 4 * s + 2 * half;
        v2f b;
        b.x = vb[(long)k * D_D];
        b.y = vb[(long)(k + 1) * D_D];
        acc = __builtin_amdgcn_wmma_f32_16x16x4_f32(
            false, a, false, b, (short)0, acc, false, false);
    }

    if (pass == 0) {
        // raw p1 in padded [slice][1024][64] layout; pad rows written as zero
#pragma unroll
        for (int r = 0; r < 8; ++r) {
            const int hr = h0 + r + 8 * half;
            float val = acc[r] * rowinv[r + 8 * half];
            Out[slice + (long)hr * D_D + n0 + l] = (hr < H_P) ? val : 0.0f;
        }
    } else {
        // tanh + transpose to (n, h, f, d)
#pragma unroll
        for (int r = 0; r < 8; ++r) {
            const int hr = h0 + r + 8 * half;
            if (hr < H_P) {
                float val = tanhf(acc[r] * rowinv[r + 8 * half]);
                Out[(((long)n * H_P + hr) * F_F + f) * D_D + n0 + l] = val;
            }
        }
    }
}

// ---------------------------------------------------------------------------
// out1 = tanh(p1) transposed (n,f,h,d) -> (n,h,f,d)
// ---------------------------------------------------------------------------
__global__ __launch_bounds__(256)
void tanh_out1_kernel(const float* __restrict__ P1, float* __restrict__ O1) {
    int idx = blockIdx.x * blockDim.x + threadIdx.x;  // N*F*H_PAD*D threads
    int d  = idx & 63;
    int h  = (idx >> 6) & 1023;
    int fz = idx >> 16;                               // n*32 + f
    if (h >= H_P) return;
    int f = fz & 31;
    int n = fz >> 5;
    O1[(((long)n * H_P + h) * F_F + f) * D_D + d] = tanhf(P1[idx]);
}

// ---------------------------------------------------------------------------
extern "C" void kernel_launch(void* const* d_in, const int* in_sizes, int n_in,
                              void* d_out, int out_size, void* d_ws, size_t ws_size,
                              hipStream_t stream) {
    (void)in_sizes; (void)n_in; (void)out_size; (void)ws_size;
    const float* prot1 = (const float*)d_in[0];
    const float* prot2 = (const float*)d_in[1];
    const float* W     = (const float*)d_in[2];
    float* out = (float*)d_out;
    float* ws  = (float*)d_ws;

    const long QSZ = (long)N_B * F_F * H_PAD * D_D;   // 8,388,608 floats
    float* Wt = ws;                                    // 114,688 floats
    float* q1 = ws + 114688;
    float* q2 = q1 + QSZ;
    float* p1 = q2 + QSZ;
    float* out1 = out;
    float* out2 = out + (long)N_B * H_P * F_F * D_D;   // 8,339,456 floats

    repack_W_kernel<<<(F_F * C_C * D_D * W_W + 255) / 256, 256, 0, stream>>>(W, Wt);

    const int projThreads = N_B * H_IN * D_D;          // 262,144
    project_kernel<<<projThreads / 256, 256, 0, stream>>>(prot1, Wt, q1);
    project_kernel<<<projThreads / 256, 256, 0, stream>>>(prot2, Wt, q2);

    const size_t ldsBytes = (size_t)(16 * SSTRIDE + 128 + 16 + 16) * sizeof(float);
    dim3 grid(64, F_F, N_B);                           // h-block, f, n
    attn_kernel<<<grid, 128, ldsBytes, stream>>>(q1, q2, q2, p1, 0);
    attn_kernel<<<grid, 128, ldsBytes, stream>>>(q1, q2, p1, out2, 1);

    tanh_out1_kernel<<<(int)(QSZ / 256), 256, 0, stream>>>(p1, out1);
}